// YoloLoss_21869973471614
// MI455X (gfx1250) — compile-verified
//
#include <hip/hip_runtime.h>

// ---------------------------------------------------------------------------
// YOLO loss on MI455X (gfx1250): memory-bound streaming reduction.
// WMMA f32 16x16x4 (full-precision matrix op) is used as the intra-wave
// reduction primitive; deterministic two-pass block reduction.
// ---------------------------------------------------------------------------

typedef float v2f __attribute__((ext_vector_type(2)));
typedef float v8f __attribute__((ext_vector_type(8)));

#define GRID_INV      (1.0f / 7.0f)
#define LAMBDA_COORD  5.0f
#define LAMBDA_NOOBJ  0.5f

// Exact f32 wave(32) sum via V_WMMA_F32_16X16X4_F32.
// A (16x4 f32): lane L puts its partial at K = (L<16 ? 0 : 2), zeros elsewhere.
// B (4x16 f32) = all ones  =>  D[m][n] = partial[m] + partial[m+16].
// D layout: lanes 0-15 hold rows M=0..7 (one per VGPR), lanes 16-31 rows 8..15.
// sum of the 8 accumulators + xor-16 shuffle = full 32-lane sum (all lanes).
// Requires EXEC == all ones (callers guarantee no divergence here).
__device__ __forceinline__ float wave_sum32(float x) {
    v2f a; a.x = x;     a.y = 0.0f;
    v2f b; b.x = 1.0f;  b.y = 1.0f;
    v8f c = {};
    c = __builtin_amdgcn_wmma_f32_16x16x4_f32(
            /*neg_a=*/false, a, /*neg_b=*/false, b,
            /*c_mod=*/(short)0, c, /*reuse_a=*/false, /*reuse_b=*/false);
    float s = c[0] + c[1] + c[2] + c[3] + c[4] + c[5] + c[6] + c[7];
    s += __shfl_xor(s, 16, 32);
    return s;
}

// Per-cell YOLO loss contribution (26 floats each of pred/target).
__device__ __forceinline__ float cell_loss(const float* __restrict__ o,
                                           const float* __restrict__ t) {
    // Rows are 104 B -> always 8-byte aligned: load as 13 float2 (global_load_b64).
    float ov[26], tv[26];
#pragma unroll
    for (int i = 0; i < 13; ++i) {
        float2 vo = ((const float2*)o)[i];
        float2 vt = ((const float2*)t)[i];
        ov[2 * i] = vo.x; ov[2 * i + 1] = vo.y;
        tv[2 * i] = vt.x; tv[2 * i + 1] = vt.y;
    }

    const float coord = (tv[4] > 0.0f)  ? 1.0f : 0.0f;
    const float noobj = (tv[4] == 0.0f) ? 1.0f : 0.0f;

    // Class loss: channels 10..25.
    float class_l = 0.0f;
#pragma unroll
    for (int c = 10; c < 26; ++c) {
        float d = ov[c] - tv[c];
        class_l += d * d;
    }

    // No-object loss: both box confidences.
    float d4 = ov[4] - tv[4];
    float d9 = ov[9] - tv[9];
    float noobj_l = d4 * d4 + d9 * d9;

    // Per-box IoU / location.
    float iou[2], pc[2], loc[2];
#pragma unroll
    for (int bidx = 0; bidx < 2; ++bidx) {
        const int s = 5 * bidx;
        float px = ov[s], py = ov[s + 1], pw = ov[s + 2], ph = ov[s + 3];
        float tx = tv[s], ty = tv[s + 1], tw = tv[s + 2], th = tv[s + 3];
        pc[bidx] = ov[s + 4];

        float pcx = px * GRID_INV, pcy = py * GRID_INV;
        float tcx = tx * GRID_INV, tcy = ty * GRID_INV;
        float phw = 0.5f * pw, phh = 0.5f * ph;
        float thw = 0.5f * tw, thh = 0.5f * th;

        float p_l = pcx - phw, p_t = pcy - phh, p_r = pcx + phw, p_b = pcy + phh;
        float t_l = tcx - thw, t_t = tcy - thh, t_r = tcx + thw, t_b = tcy + thh;

        float lx = fmaxf(p_l, t_l), ly = fmaxf(p_t, t_t);
        float rx = fminf(p_r, t_r), ry = fminf(p_b, t_b);
        float wx = fmaxf(rx - lx, 0.0f);
        float wy = fmaxf(ry - ly, 0.0f);
        float inter  = wx * wy;
        float area_p = (p_r - p_l) * (p_b - p_t);
        float area_t = (t_r - t_l) * (t_b - t_t);
        iou[bidx] = inter / (area_p + area_t - inter);

        float dx = px - tx, dy = py - ty;
        float dw = __builtin_sqrtf(pw) - __builtin_sqrtf(tw);
        float dh = __builtin_sqrtf(ph) - __builtin_sqrtf(th);
        loc[bidx] = dx * dx + dy * dy + dw * dw + dh * dh;
    }

    // argmax with first-index tie rule (matches jnp.argmax).
    const int   r       = (iou[1] > iou[0]) ? 1 : 0;
    const float iou_max = fmaxf(iou[0], iou[1]);

    float dr      = pc[r] - iou_max;
    float resp_l  = dr * dr;                // response (contain) loss
    float loc_l   = loc[r];                 // location loss, responsible box
    float nr      = pc[1 - r];
    float nresp_l = nr * nr;                // not-responsible conf loss

    return coord * (LAMBDA_COORD * loc_l + class_l + resp_l + LAMBDA_NOOBJ * nresp_l)
         + LAMBDA_NOOBJ * noobj * noobj_l;
}

// Deterministic in-block reduction: WMMA wave sum -> LDS -> thread-0 ordered sum.
__device__ __forceinline__ void block_reduce_store(float partial,
                                                   float* __restrict__ dst,
                                                   float scale) {
    const float w = wave_sum32(partial);    // all 256 threads reach this (EXEC all-1s)
    __shared__ float lds[8];
    const int lane = threadIdx.x & 31;
    const int wv   = threadIdx.x >> 5;
    if (lane == 0) lds[wv] = w;
    __syncthreads();
    if (threadIdx.x == 0) {
        float s = 0.0f;
#pragma unroll
        for (int i = 0; i < 8; ++i) s += lds[i];
        *dst = s * scale;
    }
}

__global__ void __launch_bounds__(256)
yolo_partial_kernel(const float* __restrict__ out, const float* __restrict__ tgt,
                    float* __restrict__ blocksums, int ncells) {
    const int idx = blockIdx.x * 256 + threadIdx.x;
    float partial = 0.0f;
    if (idx < ncells) {
        const size_t off = (size_t)idx * 26u;
        partial = cell_loss(out + off, tgt + off);
    }
    block_reduce_store(partial, &blocksums[blockIdx.x], 1.0f);
}

__global__ void __launch_bounds__(256)
yolo_final_kernel(const float* __restrict__ blocksums, int n, float scale,
                  float* __restrict__ result) {
    float s = 0.0f;
    for (int i = threadIdx.x; i < n; i += 256) s += blocksums[i];  // fixed order
    block_reduce_store(s, result, scale);
}

extern "C" void kernel_launch(void* const* d_in, const int* in_sizes, int n_in,
                              void* d_out, int out_size, void* d_ws, size_t ws_size,
                              hipStream_t stream) {
    const float* out = (const float*)d_in[0];   // [B,7,7,26] f32
    const float* tgt = (const float*)d_in[1];   // [B,7,7,26] f32

    const int total  = in_sizes[0];
    const int ncells = total / 26;              // B*49
    const int B      = ncells / 49;
    const int nblk   = (ncells + 255) / 256;    // 3136 for B=16384

    float* bsums = (float*)d_ws;                // nblk floats of scratch

    yolo_partial_kernel<<<nblk, 256, 0, stream>>>(out, tgt, bsums, ncells);
    yolo_final_kernel<<<1, 256, 0, stream>>>(bsums, nblk, 1.0f / (float)B,
                                             (float*)d_out);
}